// ODEINDLayer_7378753815041
// MI455X (gfx1250) — compile-verified
//
#include <hip/hip_runtime.h>
#include <hip/hip_bf16.h>

// ---- problem constants (match reference) ----
#define BS     16
#define NDIM   16
#define NSYS   (BS * NDIM)     // 256 independent systems
#define NSTEP  128
#define PORD   3               // P = ORDER+1
#define NN     (NSTEP * PORD)  // 384 unknowns per system
#define NB     (NN / 16)       // 24 blocks of 16
#define HB     5               // half bandwidth of G = A^T A

typedef __attribute__((ext_vector_type(2))) float v2f;
typedef __attribute__((ext_vector_type(8))) float v8f;

// symmetric banded f64 accessor: band[p*6 + (q-p)], 0 <= q-p <= 5
static __device__ __forceinline__ double g64(const double* band, int p, int q) {
    if (p > q) { int s = p; p = q; q = s; }
    int d = q - p;
    return (d > HB) ? 0.0 : band[p * 6 + d];
}

// 16x16x16 f32 product via 4 chained V_WMMA_F32_16X16X4_F32.
// Out = signOut * (A(^T?) * B + (Cmat ? Cmat : 0))   [sign applied after the add]
// All matrices are 16x16 row-major f32 in LDS. One full wave32 must execute this.
static __device__ __forceinline__ void mm16_store(const float* Amat, bool transA,
                                                  const float* Bmat, const float* Cmat,
                                                  float* Out, float signOut) {
    const int lane = (int)(threadIdx.x & 31);
    const int half = lane >> 4;     // 0: lanes 0-15, 1: lanes 16-31
    const int mn   = lane & 15;     // M for A-frag, N for B/C/D frags
    v8f c;
#pragma unroll
    for (int v = 0; v < 8; ++v)
        c[v] = Cmat ? Cmat[(v + 8 * half) * 16 + mn] : 0.0f;
#pragma unroll
    for (int k0 = 0; k0 < 16; k0 += 4) {
        v2f a, b;
        const int ka = k0 + 2 * half;           // A: VGPR0 -> K=k0 / k0+2 ; VGPR1 -> K=k0+1 / k0+3
        a.x = transA ? Amat[(ka + 0) * 16 + mn] : Amat[mn * 16 + (ka + 0)];
        a.y = transA ? Amat[(ka + 1) * 16 + mn] : Amat[mn * 16 + (ka + 1)];
        const int kb = k0 + 2 * half;           // B: same VGPR/half striping over K rows
        b.x = Bmat[(kb + 0) * 16 + mn];
        b.y = Bmat[(kb + 1) * 16 + mn];
        c = __builtin_amdgcn_wmma_f32_16x16x4_f32(
            /*neg_a=*/false, a, /*neg_b=*/false, b,
            /*c_mod=*/(short)0, c, /*reuse_a=*/false, /*reuse_b=*/false);
    }
#pragma unroll
    for (int v = 0; v < 8; ++v)
        Out[(v + 8 * half) * 16 + mn] = signOut * c[v];
}

// Gauss-Jordan inverse of a 16x16 f32 matrix (SPD, no pivoting).
// Column-per-lane: lanes 0-15 hold S columns, lanes 16-31 hold identity -> inverse.
static __device__ __forceinline__ void gj_invert16(const float* S, float* F) {
    const int lane = (int)(threadIdx.x & 31);
    float col[16];
#pragma unroll
    for (int r = 0; r < 16; ++r)
        col[r] = (lane < 16) ? S[r * 16 + lane] : ((r == lane - 16) ? 1.0f : 0.0f);
#pragma unroll
    for (int k = 0; k < 16; ++k) {
        float piv = __shfl(col[k], k);
        float inv = 1.0f / piv;
        col[k] *= inv;
#pragma unroll
        for (int r = 0; r < 16; ++r) {
            if (r == k) continue;
            float m = __shfl(col[r], k);
            col[r] -= m * col[k];
        }
    }
    if (lane >= 16) {
#pragma unroll
        for (int r = 0; r < 16; ++r)
            F[r * 16 + (lane - 16)] = col[r];
    }
}

// out16[r] = sum_c M(^T?)[r][c] * vec[c], wave32 cooperative, result written by lanes 0-15.
static __device__ __forceinline__ void matvec16(const float* M, bool trans,
                                                const float* vec, float* out16) {
    const int lane = (int)(threadIdx.x & 31);
    const int r = lane & 15;
    const int cb = 8 * (lane >> 4);
    float s = 0.0f;
#pragma unroll
    for (int j = 0; j < 8; ++j) {
        int c = cb + j;
        s += (trans ? M[c * 16 + r] : M[r * 16 + c]) * vec[c];
    }
    s += __shfl_xor(s, 16);
    if (lane < 16) out16[r] = s;
}

// block-tridiagonal solve using stored F (=inv(S_i)) and B blocks (f32).
static __device__ void solve_bt(const float* Fblk, const float* Bblk,
                                const float* in, float* out,
                                float* yv, float* tv, float* t2) {
    const int lane = (int)(threadIdx.x & 31);
    // forward: y_0 = w_0 ; y_i = w_i - B^T_{i-1} F_{i-1} y_{i-1}
    if (lane < 16) yv[lane] = in[lane];
    __syncthreads();
    for (int i = 1; i < NB; ++i) {
        matvec16(Fblk + (i - 1) * 256, false, yv + (i - 1) * 16, tv);
        __syncthreads();
        matvec16(Bblk + (i - 1) * 256, true, tv, t2);
        __syncthreads();
        if (lane < 16) yv[i * 16 + lane] = in[i * 16 + lane] - t2[lane];
        __syncthreads();
    }
    // back: x_23 = F_23 y_23 ; x_i = F_i (y_i - B_i x_{i+1})
    matvec16(Fblk + (NB - 1) * 256, false, yv + (NB - 1) * 16, tv);
    __syncthreads();
    if (lane < 16) out[(NB - 1) * 16 + lane] = tv[lane];
    __syncthreads();
    for (int i = NB - 2; i >= 0; --i) {
        matvec16(Bblk + i * 256, false, out + (i + 1) * 16, tv);
        __syncthreads();
        if (lane < 16) t2[lane] = yv[i * 16 + lane] - tv[lane];
        __syncthreads();
        matvec16(Fblk + i * 256, false, t2, tv);
        __syncthreads();
        if (lane < 16) out[i * 16 + lane] = tv[lane];
        __syncthreads();
    }
}

__global__ __launch_bounds__(32) void ode_banded_wmma_kernel(
    const float* __restrict__ g_coeffs,   // (NSYS, NSTEP, P)
    const float* __restrict__ g_rhs,      // (NSYS, NSTEP)
    const float* __restrict__ g_iv,       // (NSYS, 2)
    const float* __restrict__ g_steps,    // (NSYS, NSTEP-1)
    float* __restrict__ d_out)            // u0|u1|u2|zero|h
{
    const int sys  = (int)blockIdx.x;
    const int lane = (int)(threadIdx.x & 31);

    extern __shared__ __align__(16) char smem_raw[];
    double* band = (double*)smem_raw;             // 384*6 = 2304 doubles
    double* atb  = band + NN * 6;                 // 384
    double* vmod = atb + NN;                      // 384
    double* rres = vmod + NN;                     // 384
    float*  Dblk = (float*)(rres + NN);           // 24*256
    float*  Bblk = Dblk + NB * 256;               // 23*256
    float*  Fblk = Bblk + (NB - 1) * 256;         // 24*256
    float*  Tbuf = Fblk + NB * 256;               // 256
    float*  wv   = Tbuf + 256;                    // 384
    float*  yv   = wv + NN;                       // 384
    float*  xv   = yv + NN;                       // 384
    float*  dv   = xv + NN;                       // 384
    float*  tv   = dv + NN;                       // 16
    float*  t2   = tv + 16;                       // 16

    const float* coeffs = g_coeffs + (size_t)sys * NSTEP * PORD;
    const float* rhs    = g_rhs    + (size_t)sys * NSTEP;
    const float* ivp    = g_iv     + (size_t)sys * 2;
    const float* steps  = g_steps  + (size_t)sys * (NSTEP - 1);

    // ---- 0: clear band, fill Atb = A^T b (only equation rows have b != 0) ----
    for (int i = lane; i < NN * 6; i += 32) band[i] = 0.0;
    for (int i = lane; i < NN; i += 32) {
        int t = i / PORD, j = i % PORD;
        atb[i] = (double)coeffs[t * PORD + j] * (double)rhs[t];
    }
    __syncthreads();

    // ---- 1: assemble banded G = A^T A with LDS f64 atomics (analytic sparsity) ----
    const int NROWS = NSTEP + 5 * (NSTEP - 1);   // 763
    for (int r = lane; r < NROWS; r += 32) {
        int    cols[4];
        double vals[4];
        int nnz;
        if (r < NSTEP) {                                  // equation row
            int t = r;
            nnz = 3;
#pragma unroll
            for (int j = 0; j < 3; ++j) { cols[j] = 3 * t + j; vals[j] = (double)coeffs[t * 3 + j]; }
        } else {
            int r2 = r - NSTEP;
            int cls = r2 / (NSTEP - 1);
            int t   = r2 % (NSTEP - 1);
            double h = (double)steps[t];
            if (cls == 0) {            // forward Taylor, order i=0
                double sc = 1.0 / (h * h * h);
                nnz = 4;
                cols[0] = 3 * t;     vals[0] = sc;
                cols[1] = 3 * t + 1; vals[1] = sc * h;
                cols[2] = 3 * t + 2; vals[2] = sc * h * h * 0.5;
                cols[3] = 3 * t + 3; vals[3] = -sc;
            } else if (cls == 1) {     // backward Taylor, order i=0
                double sc = 1.0 / (h * h * h);
                nnz = 4;
                cols[0] = 3 * t + 3; vals[0] = sc;
                cols[1] = 3 * t + 4; vals[1] = -sc * h;
                cols[2] = 3 * t + 5; vals[2] = sc * h * h * 0.5;
                cols[3] = 3 * t;     vals[3] = -sc;
            } else if (cls == 2) {     // forward, order i=1
                double sc = 1.0 / (h * h);
                nnz = 3;
                cols[0] = 3 * t + 1; vals[0] = sc;
                cols[1] = 3 * t + 2; vals[1] = sc * h;
                cols[2] = 3 * t + 4; vals[2] = -sc;
            } else if (cls == 3) {     // backward, order i=1
                double sc = 1.0 / (h * h);
                nnz = 3;
                cols[0] = 3 * t + 4; vals[0] = sc;
                cols[1] = 3 * t + 5; vals[1] = -sc * h;
                cols[2] = 3 * t + 1; vals[2] = -sc;
            } else {                   // top-order smoothness
                double sc = 1.0 / h;
                nnz = 2;
                cols[0] = 3 * t + 5; vals[0] = sc;
                cols[1] = 3 * t + 2; vals[1] = -sc;
            }
        }
        for (int a = 0; a < nnz; ++a)
            for (int b = a; b < nnz; ++b) {
                int p = cols[a], q = cols[b];
                double v = vals[a] * vals[b];
                if (p > q) { int s = p; p = q; q = s; }
                atomicAdd(&band[p * 6 + (q - p)], v);
            }
    }
    __syncthreads();

    // ---- 2: eliminate hard IV constraints exactly (x0 = iv0, x1 = iv1) ----
    for (int i = lane; i < NN; i += 32) vmod[i] = atb[i];
    __syncthreads();
    if (lane == 0) {
        double iv0 = (double)ivp[0], iv1 = (double)ivp[1];
        for (int q = 2; q <= 6; ++q)
            vmod[q] -= g64(band, 0, q) * iv0 + g64(band, 1, q) * iv1;
        for (int d = 1; d <= HB; ++d) { band[0 * 6 + d] = 0.0; band[1 * 6 + d] = 0.0; }
        band[0 * 6 + 0] = 1.0;
        band[1 * 6 + 0] = 1.0;
        vmod[0] = iv0;
        vmod[1] = iv1;
    }
    __syncthreads();

    // ---- 3: expand band (f64) into 16x16 block-tridiagonal blocks (f32) ----
    for (int i = lane; i < NB * 256; i += 32) {
        int blk = i >> 8, rc = i & 255, r = rc >> 4, c = rc & 15;
        Dblk[i] = (float)g64(band, 16 * blk + r, 16 * blk + c);
    }
    for (int i = lane; i < (NB - 1) * 256; i += 32) {
        int blk = i >> 8, rc = i & 255, r = rc >> 4, c = rc & 15;
        Bblk[i] = (float)g64(band, 16 * blk + r, 16 * (blk + 1) + c);
    }
    for (int i = lane; i < NN; i += 32) wv[i] = (float)vmod[i];
    __syncthreads();

    // ---- 4: block-tridiagonal factorization, Schur updates on the matrix cores ----
    // S_0 = D_0 ;  F_i = inv(S_i) ;  S_{i+1} = D_{i+1} - B_i^T F_i B_i
    for (int i = 0; i < NB; ++i) {
        gj_invert16(Dblk + i * 256, Fblk + i * 256);
        __syncthreads();
        if (i < NB - 1) {
            // Tbuf = -(F_i * B_i)            (4x v_wmma_f32_16x16x4_f32)
            mm16_store(Fblk + i * 256, false, Bblk + i * 256, nullptr, Tbuf, -1.0f);
            __syncthreads();
            // D_{i+1} += B_i^T * Tbuf        (4x v_wmma, C-accumulate)
            mm16_store(Bblk + i * 256, true, Tbuf, Dblk + (i + 1) * 256,
                       Dblk + (i + 1) * 256, 1.0f);
            __syncthreads();
        }
    }

    // ---- 5: f32 solve, then one f64 iterative-refinement pass ----
    solve_bt(Fblk, Bblk, wv, xv, yv, tv, t2);
    __syncthreads();

    for (int p = lane; p < NN; p += 32) {       // f64 band residual r = v' - G' x
        double s = vmod[p];
        int q0 = p - HB < 0 ? 0 : p - HB;
        int q1 = p + HB >= NN ? NN - 1 : p + HB;
        for (int q = q0; q <= q1; ++q)
            s -= g64(band, p, q) * (double)xv[q];
        rres[p] = s;
    }
    __syncthreads();
    for (int i = lane; i < NN; i += 32) wv[i] = (float)rres[i];
    __syncthreads();
    solve_bt(Fblk, Bblk, wv, dv, yv, tv, t2);
    __syncthreads();

    // ---- 6: outputs: u0 | u1 | u2 | scalar 0 | h ----
    for (int t = lane; t < NSTEP; t += 32) {
        float u0 = xv[3 * t + 0] + dv[3 * t + 0];
        float u1 = xv[3 * t + 1] + dv[3 * t + 1];
        float u2 = xv[3 * t + 2] + dv[3 * t + 2];
        d_out[0 * NSYS * NSTEP + sys * NSTEP + t] = u0;
        d_out[1 * NSYS * NSTEP + sys * NSTEP + t] = u1;
        d_out[2 * NSYS * NSTEP + sys * NSTEP + t] = u2;
    }
    const int HOFF = 3 * NSYS * NSTEP + 1;
    for (int t = lane; t < NSTEP - 1; t += 32)
        d_out[HOFF + sys * (NSTEP - 1) + t] = steps[t];
    if (sys == 0 && lane == 0)
        d_out[3 * NSYS * NSTEP] = 0.0f;
}

extern "C" void kernel_launch(void* const* d_in, const int* in_sizes, int n_in,
                              void* d_out, int out_size, void* d_ws, size_t ws_size,
                              hipStream_t stream) {
    (void)in_sizes; (void)n_in; (void)d_ws; (void)ws_size; (void)out_size;
    const float* coeffs = (const float*)d_in[0];
    const float* rhs    = (const float*)d_in[1];
    const float* iv     = (const float*)d_in[2];
    const float* steps  = (const float*)d_in[3];

    constexpr size_t SMEM =
        (size_t)(NN * 6 + 3 * NN) * sizeof(double) +                 // band + atb + vmod + rres
        (size_t)(NB * 256 + (NB - 1) * 256 + NB * 256 + 256 +        // D, B, F, T
                 4 * NN + 32) * sizeof(float);                       // wv, yv, xv, dv, tv, t2

    ode_banded_wmma_kernel<<<NSYS, 32, SMEM, stream>>>(
        coeffs, rhs, iv, steps, (float*)d_out);
}